// GIN_25383256719665
// MI455X (gfx1250) — compile-verified
//
#include <hip/hip_runtime.h>
#include <hip/hip_bf16.h>

// ---------------------------------------------------------------------------
// GIN forward for MI455X (gfx1250, wave32).
// Matmuls use native-FP32 WMMA: V_WMMA_F32_16X16X4_F32 (D = A(16x4)*B(4x16)+C).
// Edge aggregation / pooling via L2-resident global f32 atomics.
// ---------------------------------------------------------------------------

#define N_NODES 100000
#define N_EDGES 1600000
#define N_GRAPH 512
#define DIN     78
#define DF      32
#define DOUT    128
#define BN_EPS  1e-5f

typedef __attribute__((ext_vector_type(2))) float v2f;
typedef __attribute__((ext_vector_type(8))) float v8f;

static __device__ __forceinline__ v8f wmma4(v2f a, v2f b, v8f c) {
    // 8 args: (neg_a, A, neg_b, B, c_mod, C, reuse_a, reuse_b)
    return __builtin_amdgcn_wmma_f32_16x16x4_f32(
        false, a, false, b, (short)0, c, false, false);
}

// ---------------------------------------------------------------------------
// Zero fill
// ---------------------------------------------------------------------------
__global__ void fill_zero(float* __restrict__ p, int n) {
    int i = blockIdx.x * blockDim.x + threadIdx.x;
    if (i < n) p[i] = 0.0f;
}

// ---------------------------------------------------------------------------
// Edge aggregation: agg[dst] += h[src].  One wave per edge, lane = feature.
// ---------------------------------------------------------------------------
__global__ void aggregate(const float* __restrict__ h,
                          const int* __restrict__ edge_index,
                          float* __restrict__ agg, int nEdges, int D) {
    int e    = blockIdx.x * (blockDim.x >> 5) + (threadIdx.x >> 5);
    int lane = threadIdx.x & 31;
    if (e >= nEdges) return;
    int s = edge_index[e];           // row 0 = src
    int d = edge_index[nEdges + e];  // row 1 = dst
    for (int f = lane; f < D; f += 32)
        atomicAdd(&agg[(size_t)d * D + f], h[(size_t)s * D + f]);
}

// ---------------------------------------------------------------------------
// GIN conv: out = relu( relu((h+agg) @ W1 + b1) @ W2 + b2 )
// plus per-feature sum / sum-of-squares accumulation for BatchNorm.
// One wave handles a 16-node tile.  KPAD = K padded to mult of 4,
// TS = LDS tile row stride (>= 33).
// ---------------------------------------------------------------------------
template<int KIN, int KPAD, int TS>
__global__ __launch_bounds__(256) void conv_kernel(
    const float* __restrict__ hin, const float* __restrict__ agg,
    const float* __restrict__ w1,  const float* __restrict__ b1,
    const float* __restrict__ w2,  const float* __restrict__ b2,
    float* __restrict__ hout,
    float* __restrict__ bnSum, float* __restrict__ bnSq, int nTiles)
{
    __shared__ float sW1[KPAD * DF];
    __shared__ float sW2[DF * DF];
    __shared__ float sB1[DF];
    __shared__ float sB2[DF];
    __shared__ float sTile[8][16 * TS];
    __shared__ float sWS[8][DF];
    __shared__ float sWQ[8][DF];

    int tid = threadIdx.x;
    for (int i = tid; i < KPAD * DF; i += 256)
        sW1[i] = (i < KIN * DF) ? w1[i] : 0.0f;
    for (int i = tid; i < DF * DF; i += 256)
        sW2[i] = w2[i];
    if (tid < DF) { sB1[tid] = b1[tid]; sB2[tid] = b2[tid]; }
    __syncthreads();

    int wave = tid >> 5;
    int lane = tid & 31;
    int tile = blockIdx.x * 8 + wave;
    bool active = tile < nTiles;          // uniform per wave

    float s0 = 0.f, s1 = 0.f, q0 = 0.f, q1 = 0.f;

    if (active) {
        float* T  = sTile[wave];
        int base  = tile * 16;

        // load z = h + agg tile (16 x KIN), zero-pad to KPAD
        for (int r = 0; r < 16; ++r) {
            size_t row = (size_t)(base + r) * KIN;
            for (int f = lane; f < KPAD; f += 32)
                T[r * TS + f] = (f < KIN) ? (hin[row + f] + agg[row + f]) : 0.0f;
        }
        __builtin_amdgcn_wave_barrier();

        const int m    = lane & 15;                // A row
        const int koff = (lane >> 4) << 1;         // A/B k sub-offset
        const int n    = lane & 15;                // B/C column within tile
        const int Mb   = (lane >> 4) << 3;         // C row base

        // ---- matmul 1: (16 x KPAD) @ (KPAD x 32) ----
        v8f acc0 = {}; v8f acc1 = {};
        for (int k0 = 0; k0 < KPAD; k0 += 4) {
            int kb = k0 + koff;
            v2f a  = { T[m * TS + kb], T[m * TS + kb + 1] };
            v2f bA = { sW1[kb * DF + n],       sW1[(kb + 1) * DF + n] };
            v2f bB = { sW1[kb * DF + 16 + n],  sW1[(kb + 1) * DF + 16 + n] };
            acc0 = wmma4(a, bA, acc0);
            acc1 = wmma4(a, bB, acc1);
        }

        // bias + relu, stash H1 back into LDS (A-layout source, stride 33)
        for (int v = 0; v < 8; ++v) {
            float x0 = fmaxf(acc0[v] + sB1[n],      0.0f);
            float x1 = fmaxf(acc1[v] + sB1[16 + n], 0.0f);
            T[(Mb + v) * 33 + n]      = x0;
            T[(Mb + v) * 33 + 16 + n] = x1;
        }
        __builtin_amdgcn_wave_barrier();

        // ---- matmul 2: (16 x 32) @ (32 x 32) ----
        acc0 = (v8f){}; acc1 = (v8f){};
        for (int k0 = 0; k0 < DF; k0 += 4) {
            int kb = k0 + koff;
            v2f a  = { T[m * 33 + kb], T[m * 33 + kb + 1] };
            v2f bA = { sW2[kb * DF + n],       sW2[(kb + 1) * DF + n] };
            v2f bB = { sW2[kb * DF + 16 + n],  sW2[(kb + 1) * DF + 16 + n] };
            acc0 = wmma4(a, bA, acc0);
            acc1 = wmma4(a, bB, acc1);
        }
        __builtin_amdgcn_wave_barrier();

        // bias + outer relu, stats, stash into LDS for coalesced store
        for (int v = 0; v < 8; ++v) {
            float x0 = fmaxf(acc0[v] + sB2[n],      0.0f);
            float x1 = fmaxf(acc1[v] + sB2[16 + n], 0.0f);
            s0 += x0; q0 += x0 * x0;
            s1 += x1; q1 += x1 * x1;
            T[(Mb + v) * 33 + n]      = x0;
            T[(Mb + v) * 33 + 16 + n] = x1;
        }
        __builtin_amdgcn_wave_barrier();

        for (int r = 0; r < 16; ++r)
            hout[(size_t)(base + r) * DF + lane] = T[r * 33 + lane];
    }

    // per-wave feature reduction (lane & lane+16 hold the same feature halves)
    s0 += __shfl_xor(s0, 16, 32);  q0 += __shfl_xor(q0, 16, 32);
    s1 += __shfl_xor(s1, 16, 32);  q1 += __shfl_xor(q1, 16, 32);
    if (lane < 16) {
        sWS[wave][lane]      = s0;  sWS[wave][16 + lane] = s1;
        sWQ[wave][lane]      = q0;  sWQ[wave][16 + lane] = q1;
    }
    __syncthreads();

    if (tid < DF) {
        float ts = 0.f, tq = 0.f;
        for (int w = 0; w < 8; ++w) { ts += sWS[w][tid]; tq += sWQ[w][tid]; }
        atomicAdd(&bnSum[tid], ts);
        atomicAdd(&bnSq[tid],  tq);
    }
}

// ---------------------------------------------------------------------------
// BatchNorm: fold stats into per-feature scale/shift
// ---------------------------------------------------------------------------
__global__ void bn_finalize(const float* __restrict__ sum,
                            const float* __restrict__ sq,
                            const float* __restrict__ gamma,
                            const float* __restrict__ beta,
                            float* __restrict__ scale,
                            float* __restrict__ shift, float invN) {
    int f = threadIdx.x;
    if (f < DF) {
        float mu  = sum[f] * invN;
        float var = sq[f] * invN - mu * mu;
        float sc  = gamma[f] * rsqrtf(var + BN_EPS);
        scale[f] = sc;
        shift[f] = beta[f] - mu * sc;
    }
}

__global__ void bn_apply(const float* __restrict__ in,
                         const float* __restrict__ scale,
                         const float* __restrict__ shift,
                         float* __restrict__ out, int n) {
    int i = blockIdx.x * blockDim.x + threadIdx.x;
    if (i < n) {
        int f = i & (DF - 1);
        out[i] = fmaf(in[i], scale[f], shift[f]);
    }
}

// ---------------------------------------------------------------------------
// global_add_pool: pooled[batch[node]] += h[node].  Wave per node.
// ---------------------------------------------------------------------------
__global__ void pool_kernel(const float* __restrict__ h,
                            const int* __restrict__ batch,
                            float* __restrict__ pooled, int nNodes) {
    int node = blockIdx.x * (blockDim.x >> 5) + (threadIdx.x >> 5);
    int lane = threadIdx.x & 31;
    if (node >= nNodes) return;
    int g = batch[node];
    atomicAdd(&pooled[(size_t)g * DF + lane], h[(size_t)node * DF + lane]);
}

// ---------------------------------------------------------------------------
// Final FC: out = relu(pooled @ fc_w + fc_b)   [512x32]@[32x128]
// Block = one 16-row tile of pooled; wave = one 16-col tile of output.
// ---------------------------------------------------------------------------
__global__ __launch_bounds__(256) void fc_wmma(
    const float* __restrict__ pooled, const float* __restrict__ fcw,
    const float* __restrict__ fcb, float* __restrict__ out) {
    __shared__ float sA[16 * 33];
    __shared__ float sW[DF * DOUT];
    __shared__ float sB[DOUT];

    int tid = threadIdx.x;
    for (int i = tid; i < DF * DOUT; i += 256) sW[i] = fcw[i];
    if (tid < DOUT) sB[tid] = fcb[tid];

    int mbase = blockIdx.x * 16;
    for (int i = tid; i < 16 * DF; i += 256) {
        int r = i >> 5, c = i & 31;
        sA[r * 33 + c] = pooled[(size_t)(mbase + r) * DF + c];
    }
    __syncthreads();

    int wave = tid >> 5, lane = tid & 31;
    int n0   = wave * 16;
    int m    = lane & 15;
    int koff = (lane >> 4) << 1;
    int n    = lane & 15;
    int Mb   = (lane >> 4) << 3;

    v8f acc = {};
    for (int k0 = 0; k0 < DF; k0 += 4) {
        int kb = k0 + koff;
        v2f a = { sA[m * 33 + kb], sA[m * 33 + kb + 1] };
        v2f b = { sW[kb * DOUT + n0 + n], sW[(kb + 1) * DOUT + n0 + n] };
        acc = wmma4(a, b, acc);
    }
    for (int v = 0; v < 8; ++v) {
        float val = fmaxf(acc[v] + sB[n0 + n], 0.0f);
        out[(size_t)(mbase + Mb + v) * DOUT + n0 + n] = val;
    }
}

// ---------------------------------------------------------------------------
// Host side
// ---------------------------------------------------------------------------
static inline int ceil_div(long long a, long long b) { return (int)((a + b - 1) / b); }

extern "C" void kernel_launch(void* const* d_in, const int* in_sizes, int n_in,
                              void* d_out, int out_size, void* d_ws, size_t ws_size,
                              hipStream_t stream) {
    const float* x      = (const float*)d_in[0];
    const int*   ei     = (const int*)  d_in[1];
    const int*   batch  = (const int*)  d_in[2];
    const float* w1_in  = (const float*)d_in[3];
    const float* b1_in  = (const float*)d_in[4];
    const float* w1_out = (const float*)d_in[5];
    const float* b1_out = (const float*)d_in[6];
    const float* wa     = (const float*)d_in[7];
    const float* ba     = (const float*)d_in[8];
    const float* wb     = (const float*)d_in[9];
    const float* bb     = (const float*)d_in[10];
    const float* gamma  = (const float*)d_in[11];
    const float* beta   = (const float*)d_in[12];
    const float* fc_w   = (const float*)d_in[13];
    const float* fc_b   = (const float*)d_in[14];
    float* out = (float*)d_out;

    float* ws      = (float*)d_ws;
    float* agg     = ws;                                  // N * 80 slots
    float* hcur    = agg     + (size_t)N_NODES * 80;      // N * 32
    float* hnext   = hcur    + (size_t)N_NODES * DF;      // N * 32
    float* pooled  = hnext   + (size_t)N_NODES * DF;      // G * 32
    float* bnStats = pooled  + (size_t)N_GRAPH * DF;      // 64 (sum | sumsq)
    float* bnScale = bnStats + 64;                        // 32
    float* bnShift = bnScale + DF;                        // 32

    const int nTiles = N_NODES / 16;            // 6250, exact
    const int convBlocks = ceil_div(nTiles, 8); // 8 waves/block
    const int edgeBlocks = ceil_div(N_EDGES, 8);
    const float invN = 1.0f / (float)N_NODES;

    // ---------------- layer 1 (DIN=78 -> 32) ----------------
    fill_zero<<<ceil_div((long long)N_NODES * DIN, 256), 256, 0, stream>>>(agg, N_NODES * DIN);
    fill_zero<<<1, 64, 0, stream>>>(bnStats, 64);
    aggregate<<<edgeBlocks, 256, 0, stream>>>(x, ei, agg, N_EDGES, DIN);
    conv_kernel<DIN, 80, 81><<<convBlocks, 256, 0, stream>>>(
        x, agg, w1_in, b1_in, w1_out, b1_out, hnext, bnStats, bnStats + 32, nTiles);
    bn_finalize<<<1, 32, 0, stream>>>(bnStats, bnStats + 32, gamma, beta,
                                      bnScale, bnShift, invN);
    bn_apply<<<ceil_div((long long)N_NODES * DF, 256), 256, 0, stream>>>(
        hnext, bnScale, bnShift, hcur, N_NODES * DF);

    // ---------------- layers 2..5 (32 -> 32) ----------------
    for (int i = 0; i < 4; ++i) {
        fill_zero<<<ceil_div((long long)N_NODES * DF, 256), 256, 0, stream>>>(agg, N_NODES * DF);
        fill_zero<<<1, 64, 0, stream>>>(bnStats, 64);
        aggregate<<<edgeBlocks, 256, 0, stream>>>(hcur, ei, agg, N_EDGES, DF);
        conv_kernel<DF, DF, 33><<<convBlocks, 256, 0, stream>>>(
            hcur, agg, wa + (size_t)i * DF * DF, ba + (size_t)i * DF,
            wb + (size_t)i * DF * DF, bb + (size_t)i * DF,
            hnext, bnStats, bnStats + 32, nTiles);
        bn_finalize<<<1, 32, 0, stream>>>(bnStats, bnStats + 32,
                                          gamma + (size_t)(i + 1) * DF,
                                          beta + (size_t)(i + 1) * DF,
                                          bnScale, bnShift, invN);
        bn_apply<<<ceil_div((long long)N_NODES * DF, 256), 256, 0, stream>>>(
            hnext, bnScale, bnShift, hcur, N_NODES * DF);
    }

    // ---------------- pool + FC ----------------
    fill_zero<<<ceil_div((long long)N_GRAPH * DF, 256), 256, 0, stream>>>(pooled, N_GRAPH * DF);
    pool_kernel<<<ceil_div(N_NODES, 8), 256, 0, stream>>>(hcur, batch, pooled, N_NODES);
    fc_wmma<<<N_GRAPH / 16, 256, 0, stream>>>(pooled, fc_w, fc_b, out);
}